// MultiKR_90494960927208
// MI455X (gfx1250) — compile-verified
//
#include <hip/hip_runtime.h>

// CDNA5 f32 WMMA fragments (wave32): A 16x4 -> 2 VGPRs, B 4x16 -> 2 VGPRs, C/D 16x16 -> 8 VGPRs
typedef float v2f __attribute__((ext_vector_type(2)));
typedef float v8f __attribute__((ext_vector_type(8)));

#define TM    16    // rows per workgroup tile (WMMA M)
#define LDA   132   // padded LDS row stride for 128-wide activations (bank-conflict-free)
#define LDH   260   // padded LDS row stride for 256-wide activations
#define WAVES 8     // waves per workgroup, cooperating on one 16-row tile

__device__ __forceinline__ float dot4(float4 a, float4 b) {
  return a.x * b.x + a.y * b.y + a.z * b.z + a.w * b.w;
}

// Accumulate acc += A_tile(16 x 4*kblks, from LDS) * W(rows k, cols n0..n0+15, from global)
__device__ __forceinline__ void mma_accum(const float* sA, int lda,
                                          const float* __restrict__ W, int ldw,
                                          int kblks, int lo, int hi, int n0, v8f& acc) {
  const int c = n0 + lo;
  #pragma unroll 4
  for (int kb = 0; kb < kblks; ++kb) {
    const int k0 = 4 * kb + 2 * hi;
    v2f a, b;
    a.x = sA[lo * lda + k0];
    a.y = sA[lo * lda + k0 + 1];
    b.x = W[(size_t)k0 * ldw + c];
    b.y = W[(size_t)(k0 + 1) * ldw + c];
    acc = __builtin_amdgcn_wmma_f32_16x16x4_f32(false, a, false, b, (short)0, acc,
                                                false, false);
  }
}

// D-layout store with bias + ReLU: VGPR r -> row (r + 8*hi), col (n0 + lo)
__device__ __forceinline__ void store_tile_relu(float* sD, int ldd,
                                                const float* __restrict__ bias,
                                                int n0, int lo, int hi, v8f acc) {
  const float bb = bias[n0 + lo];
  #pragma unroll
  for (int r = 0; r < 8; ++r) {
    float v = acc[r] + bb;
    sD[(r + 8 * hi) * ldd + (n0 + lo)] = v > 0.f ? v : 0.f;
  }
}

__global__ __launch_bounds__(WAVES * 32) void multikr_fused(
    const int* __restrict__ user_ids, const int* __restrict__ item_ids,
    const float* __restrict__ rec_target,
    const float* __restrict__ user_tbl, const float* __restrict__ item_tbl,
    const float* __restrict__ entity_tbl,
    const float* __restrict__ w_vv, const float* __restrict__ w_ev,
    const float* __restrict__ w_ve, const float* __restrict__ w_ee,
    const float* __restrict__ bias_v, const float* __restrict__ bias_e,
    const float* __restrict__ Wu, const float* __restrict__ bu,
    const float* __restrict__ W1, const float* __restrict__ b1,
    const float* __restrict__ W2, const float* __restrict__ b2,
    const float* __restrict__ W3, const float* __restrict__ b3,
    float* __restrict__ out, int Bn) {
  __shared__ float su[TM * LDA];  // user activations / h2
  __shared__ float sv[TM * LDA];  // item_embed after cross-compress
  __shared__ float sh[TM * LDH];  // ping buffer / h1 (256 cols)

  const int wv   = threadIdx.x >> 5;   // wave id 0..7
  const int lane = threadIdx.x & 31;
  const int lo   = lane & 15;
  const int hi   = lane >> 4;
  const int base = blockIdx.x * TM;    // one 16-row tile per workgroup

  // ---- Phase 1: gather user rows into LDS (2 rows per wave, 512B coalesced) ----
  #pragma unroll
  for (int i = 0; i < 2; ++i) {
    const int r   = wv * 2 + i;
    const int uid = user_ids[base + r];
    float4 t = *(const float4*)(user_tbl + (size_t)uid * 128 + lane * 4);
    *(float4*)(su + r * LDA + lane * 4) = t;
  }

  // ---- Phase 2: cross-compress x2 (2 rows per wave; factorized dots + elementwise) ----
  const float4 Wvv = *(const float4*)(w_vv + lane * 4);
  const float4 Wev = *(const float4*)(w_ev + lane * 4);
  const float4 Wve = *(const float4*)(w_ve + lane * 4);
  const float4 Wee = *(const float4*)(w_ee + lane * 4);
  const float  bv = bias_v[0], be = bias_e[0];

  #pragma unroll
  for (int i = 0; i < 2; ++i) {
    const int r   = wv * 2 + i;
    const int iid = item_ids[base + r];
    float4 v = *(const float4*)(item_tbl   + (size_t)iid * 128 + lane * 4);
    float4 e = *(const float4*)(entity_tbl + (size_t)iid * 128 + lane * 4);
    #pragma unroll
    for (int layer = 0; layer < 2; ++layer) {
      float hv = dot4(e, Wvv);  // head @ w_vv
      float iv = dot4(v, Wev);  // item @ w_ev
      float he = dot4(e, Wve);  // head @ w_ve
      float ie = dot4(v, Wee);  // item @ w_ee
      #pragma unroll
      for (int off = 16; off > 0; off >>= 1) {
        hv += __shfl_xor(hv, off, 32);
        iv += __shfl_xor(iv, off, 32);
        he += __shfl_xor(he, off, 32);
        ie += __shfl_xor(ie, off, 32);
      }
      float4 nv, ne;
      nv.x = v.x * hv + e.x * iv + bv;  ne.x = v.x * he + e.x * ie + be;
      nv.y = v.y * hv + e.y * iv + bv;  ne.y = v.y * he + e.y * ie + be;
      nv.z = v.z * hv + e.z * iv + bv;  ne.z = v.z * he + e.z * ie + be;
      nv.w = v.w * hv + e.w * iv + bv;  ne.w = v.w * he + e.w * ie + be;
      v = nv; e = ne;
    }
    *(float4*)(sv + r * LDA + lane * 4) = v;
  }
  __syncthreads();

  const v8f vzero = {0.f, 0.f, 0.f, 0.f, 0.f, 0.f, 0.f, 0.f};

  // ---- Phase 3: user MLP layer 1: su(16x128) @ Wu(128x128) -> sh[:, :128] ----
  {
    const int nt = wv;  // one 16-col output tile per wave
    v8f acc = vzero;
    mma_accum(su, LDA, Wu, 128, 32, lo, hi, nt * 16, acc);
    store_tile_relu(sh, LDH, bu, nt * 16, lo, hi, acc);
  }
  __syncthreads();

  // ---- Phase 4: user MLP layer 2: sh -> su ----
  {
    const int nt = wv;
    v8f acc = vzero;
    mma_accum(sh, LDH, Wu, 128, 32, lo, hi, nt * 16, acc);
    store_tile_relu(su, LDA, bu, nt * 16, lo, hi, acc);
  }
  __syncthreads();

  // ---- Phase 5: h1 = relu([u2 | v2](16x256) @ W1(256x256) + b1) -> sh (2 tiles/wave) ----
  #pragma unroll
  for (int i = 0; i < 2; ++i) {
    const int nt = wv * 2 + i;
    v8f acc = vzero;
    mma_accum(su, LDA, W1,             256, 32, lo, hi, nt * 16, acc);  // k = 0..127
    mma_accum(sv, LDA, W1 + 128 * 256, 256, 32, lo, hi, nt * 16, acc);  // k = 128..255
    store_tile_relu(sh, LDH, b1, nt * 16, lo, hi, acc);
  }
  __syncthreads();

  // ---- Phase 6: h2 = relu(sh(16x256) @ W2(256x128) + b2) -> su ----
  {
    const int nt = wv;
    v8f acc = vzero;
    mma_accum(sh, LDH, W2, 128, 64, lo, hi, nt * 16, acc);
    store_tile_relu(su, LDA, b2, nt * 16, lo, hi, acc);
  }
  __syncthreads();

  // ---- Phase 7: out = relu(h2 @ W3 + b3); copy rec_target (2 rows per wave) ----
  const float4 W3l = *(const float4*)(W3 + lane * 4);
  const float  b3v = b3[0];
  #pragma unroll
  for (int i = 0; i < 2; ++i) {
    const int r = wv * 2 + i;
    float4 h = *(const float4*)(su + r * LDA + lane * 4);
    float p = dot4(h, W3l);
    #pragma unroll
    for (int off = 16; off > 0; off >>= 1) p += __shfl_xor(p, off, 32);
    if (lane == 0) {
      float o = p + b3v;
      out[base + r]      = o > 0.f ? o : 0.f;
      out[Bn + base + r] = rec_target[base + r];
    }
  }
}

extern "C" void kernel_launch(void* const* d_in, const int* in_sizes, int n_in,
                              void* d_out, int out_size, void* d_ws, size_t ws_size,
                              hipStream_t stream) {
  (void)n_in; (void)out_size; (void)d_ws; (void)ws_size;
  const int*   user_ids   = (const int*)d_in[0];
  const int*   item_ids   = (const int*)d_in[1];
  const float* rec_target = (const float*)d_in[2];
  const float* user_tbl   = (const float*)d_in[3];
  const float* item_tbl   = (const float*)d_in[4];
  const float* entity_tbl = (const float*)d_in[5];
  const float* w_vv = (const float*)d_in[6];
  const float* w_ev = (const float*)d_in[7];
  const float* w_ve = (const float*)d_in[8];
  const float* w_ee = (const float*)d_in[9];
  const float* bias_v = (const float*)d_in[10];
  const float* bias_e = (const float*)d_in[11];
  const float* Wu = (const float*)d_in[12];
  const float* bu = (const float*)d_in[13];
  const float* W1 = (const float*)d_in[14];
  const float* b1 = (const float*)d_in[15];
  const float* W2 = (const float*)d_in[16];
  const float* b2 = (const float*)d_in[17];
  const float* W3 = (const float*)d_in[18];
  const float* b3 = (const float*)d_in[19];
  float* out = (float*)d_out;

  const int Bn     = in_sizes[0];  // 16384
  const int blocks = Bn / TM;      // 1024 workgroups, 8 waves each

  multikr_fused<<<dim3(blocks), dim3(WAVES * 32), 0, stream>>>(
      user_ids, item_ids, rec_target, user_tbl, item_tbl, entity_tbl,
      w_vv, w_ev, w_ve, w_ee, bias_v, bias_e,
      Wu, bu, W1, b1, W2, b2, W3, b3, out, Bn);
}